// GADBase_LST_60524679135353
// MI455X (gfx1250) — compile-verified
//
#include <hip/hip_runtime.h>
#include <stdint.h>

#define B_    4
#define H_    512
#define W_    512
#define NPRE  128
#define LAM   0.24f
#define EPSV  1e-8f

#define N_IMG   (B_ * H_ * W_)            // 1048576
#define N_CV    (B_ * (H_ - 1) * W_)      // 1046528
#define CV_PB   ((H_ - 1) * W_)           // 261632
#define CH_PB   (H_ * (W_ - 1))           // 261632

// ---------------------------------------------------------------------------
// Kernel A: conductances cv/ch from feats = [image(3ch), y_bicubic]
// ---------------------------------------------------------------------------
__global__ __launch_bounds__(256) void conduct_kernel(
    const float* __restrict__ image,   // [B][3][512][512]
    const float* __restrict__ ybic,    // [B][512][512]
    const float* __restrict__ logk,    // scalar
    float* __restrict__ cv,            // [B][511][512]
    float* __restrict__ chn)           // [B][512][511]
{
    int idx = blockIdx.x * 256 + threadIdx.x;
    if (idx >= N_IMG) return;
    int b   = idx >> 18;
    int rem = idx & ((1 << 18) - 1);
    int y   = rem >> 9;
    int x   = rem & 511;

    float K     = expf(logk[0]);
    float invK2 = 1.0f / (K * K);

    const float* im = image + (size_t)b * (3u << 18);
    const float* yb = ybic  + (size_t)b * (1u << 18);
    int p = (y << 9) | x;
    float f0 = im[p];
    float f1 = im[(1 << 18) + p];
    float f2 = im[(2 << 18) + p];
    float f3 = yb[p];

    if (y < H_ - 1) {
        int q = p + W_;
        float d = fabsf(im[q] - f0) + fabsf(im[(1 << 18) + q] - f1)
                + fabsf(im[(2 << 18) + q] - f2) + fabsf(yb[q] - f3);
        d *= 0.25f;
        cv[(size_t)b * CV_PB + (size_t)y * W_ + x] = 1.0f / (1.0f + d * d * invK2);
    }
    if (x < W_ - 1) {
        int q = p + 1;
        float d = fabsf(im[q] - f0) + fabsf(im[(1 << 18) + q] - f1)
                + fabsf(im[(2 << 18) + q] - f2) + fabsf(yb[q] - f3);
        d *= 0.25f;
        chn[(size_t)b * CH_PB + (size_t)y * (W_ - 1) + x] = 1.0f / (1.0f + d * d * invK2);
    }
}

// ---------------------------------------------------------------------------
// Kernel B: one fused diffuse + adjust iteration.
// Grid: 256 WGs = 4 batches x 64 block-rows of 8 rows. Block: 256 threads.
// ---------------------------------------------------------------------------
__global__ __launch_bounds__(256) void diffuse_adjust_kernel(
    const float* __restrict__ src,     // I in  [B][512][512]
    float* __restrict__ dst,           // I out [B][512][512]
    const float* __restrict__ cv,      // [B][511][512]
    const float* __restrict__ chn,     // [B][512][511]
    const float* __restrict__ source,  // [B][64][64]
    const float* __restrict__ mask)    // [B][64][64]
{
    __shared__ __align__(16) float T[10 * W_];   // tile + 1 halo row top/bottom
    __shared__ float part[8 * 64];               // per-(row, block) partial sums
    __shared__ float ratio[64];

    const int tid  = threadIdx.x;
    const int wg   = blockIdx.x;
    const int b    = wg >> 6;          // batch
    const int trow = wg & 63;          // block-row index (0..63)
    const int r0   = trow << 3;        // first image row of tile

    // ---- async stage: rows [r0-1 .. r0+8] clamped, 512 cols, as b128 ----
    const float* srcb = src + ((size_t)b << 18);
    for (int i = tid; i < 10 * (W_ / 4); i += 256) {   // 1280 float4's, 5 per thread
        int ly = i >> 7;               // 0..9
        int xx = (i & 127) << 2;       // float4 col start
        int gy = r0 - 1 + ly;
        gy = gy < 0 ? 0 : (gy > H_ - 1 ? H_ - 1 : gy);
        unsigned goff  = (unsigned)(((gy << 9) + xx) << 2);              // byte off in batch image
        unsigned laddr = (unsigned)(uintptr_t)(&T[(ly << 9) + xx]);      // LDS byte addr (low 32 bits)
        asm volatile("global_load_async_to_lds_b128 %0, %1, %2 offset:0"
                     :: "v"(laddr), "v"(goff), "s"(srcb) : "memory");
    }
    asm volatile("s_wait_asynccnt 0" ::: "memory");
    __syncthreads();

    // ---- diffusion stencil: thread (r,l) handles row r0+r, cols l+32k ----
    const int r = tid >> 5;            // 0..7
    const int l = tid & 31;            // lane
    const int y = r0 + r;
    const float* cvr  = cv  + (size_t)b * CV_PB + (size_t)y * W_;        // cv[y][.]
    const float* cvr1 = cvr - W_;                                        // cv[y-1][.]
    const float* chr  = chn + (size_t)b * CH_PB + (size_t)y * (W_ - 1);  // ch[y][.]
    const int lyc = (r + 1) << 9;      // LDS row offset of row y

    float vnew[16];
#pragma unroll
    for (int k = 0; k < 16; ++k) {
        int x = l + (k << 5);
        float Ic  = T[lyc + x];
        float acc = 0.0f;
        if (y < H_ - 1) acc += cvr[x]      * (T[lyc + W_ + x] - Ic);   // down edge
        if (y > 0)      acc -= cvr1[x]     * (Ic - T[lyc - W_ + x]);   // up edge
        if (x < W_ - 1) acc += chr[x]      * (T[lyc + x + 1] - Ic);    // right edge
        if (x > 0)      acc -= chr[x - 1]  * (Ic - T[lyc + x - 1]);    // left edge
        vnew[k] = Ic + LAM * acc;
    }

    // ---- 8x8 block sums: block j = (l>>3) + 4k for pixel (r, l+32k) ----
#pragma unroll
    for (int k = 0; k < 16; ++k) {
        float s = vnew[k];
        s += __shfl_xor(s, 1, 32);
        s += __shfl_xor(s, 2, 32);
        s += __shfl_xor(s, 4, 32);     // sum over aligned 8-lane group
        if ((l & 7) == 0) part[(r << 6) + (l >> 3) + (k << 2)] = s;
    }
    __syncthreads();

    if (tid < 64) {
        float s = 0.0f;
#pragma unroll
        for (int rr = 0; rr < 8; ++rr) s += part[(rr << 6) + tid];
        float mean = s * (1.0f / 64.0f);
        int sidx = (b << 12) + (trow << 6) + tid;      // [B][64][64]
        float m  = mask[sidx];
        ratio[tid] = (m < 0.5f) ? 1.0f : (source[sidx] / (mean + EPSV));
    }
    __syncthreads();

    // ---- scale + store (coalesced) ----
    float* dstb = dst + ((size_t)b << 18) + ((size_t)y << 9);
#pragma unroll
    for (int k = 0; k < 16; ++k) {
        int x = l + (k << 5);
        dstb[x] = vnew[k] * ratio[(l >> 3) + (k << 2)];
    }
}

// ---------------------------------------------------------------------------
extern "C" void kernel_launch(void* const* d_in, const int* in_sizes, int n_in,
                              void* d_out, int out_size, void* d_ws, size_t ws_size,
                              hipStream_t stream) {
    const float* image  = (const float*)d_in[0];
    const float* source = (const float*)d_in[1];
    const float* mask   = (const float*)d_in[2];
    const float* ybic   = (const float*)d_in[3];
    const float* logk   = (const float*)d_in[4];

    float* out = (float*)d_out;
    float* cv  = out + N_IMG;          // [4,511,512]
    float* chn = out + N_IMG + N_CV;   // [4,512,511]

    float* ping = (float*)d_ws;        // 4 MiB
    float* pong = ping + N_IMG;        // 4 MiB

    conduct_kernel<<<(N_IMG + 255) / 256, 256, 0, stream>>>(image, ybic, logk, cv, chn);

    const float* cur = ybic;           // iter 0 reads the input directly (never mutated)
    for (int it = 0; it < NPRE; ++it) {
        float* dstp = (it == NPRE - 1) ? out : ((it & 1) ? pong : ping);
        diffuse_adjust_kernel<<<B_ * (H_ / 8), 256, 0, stream>>>(cur, dstp, cv, chn, source, mask);
        cur = dstp;
    }
}